// RegionLoss_43456479101509
// MI455X (gfx1250) — compile-verified
//
#include <hip/hip_runtime.h>
#include <cstdint>

#define NB 128
#define NA 5
#define NC 80
#define NH 28
#define NW 28
#define NT 50
#define CH (5 + NC)          // 85 channels per anchor
#define PLANE (NH * NW)      // 784
#define CELLS_PER_B (NA * PLANE)  // 3920
#define RECF 16              // floats per GT record
#define OBJECT_SCALE 5.0f
#define SIL_THRESH 0.6f

typedef float v4f __attribute__((ext_vector_type(4)));
typedef float v2f __attribute__((ext_vector_type(2)));

__constant__ float c_aw[NA] = {2.8f, 5.1f, 6.4f, 9.2f, 14.1f};
__constant__ float c_ah[NA] = {4.8f, 10.2f, 16.3f, 19.8f, 23.7f};

__device__ __forceinline__ float sigmoidf_fast(float v) {
    return __fdividef(1.0f, 1.0f + __expf(-v));
}

// Record layout (16 floats): 0 xl, 1 xr, 2 yl, 3 yr, 4 area, 5 cell(int bits),
//                            6 tx, 7 ty, 8 tw, 9 th, 10 tconf, 11 tcls, 12-15 pad
__global__ void build_gt_kernel(const float* __restrict__ out,
                                const float* __restrict__ target,
                                float* __restrict__ rec) {
    int b = blockIdx.x;
    int t = threadIdx.x;
    if (t >= NT) return;

    const float* tg = target + (b * NT + t) * 5;
    float cls = tg[0];
    float gx = tg[1] * (float)NW;
    float gy = tg[2] * (float)NH;
    float gw = tg[3] * (float)NW;
    float gh = tg[4] * (float)NH;
    float area_g = gw * gh;

    // best anchor by wh-IoU (centered boxes: inter = min(w)*min(h)); first max wins
    int bn = 0;
    float best = -1.0f;
#pragma unroll
    for (int a = 0; a < NA; ++a) {
        float inter = fminf(gw, c_aw[a]) * fminf(gh, c_ah[a]);
        float uni = area_g + c_aw[a] * c_ah[a] - inter;
        float iou = __fdividef(inter, uni);
        if (iou > best) { best = iou; bn = a; }
    }
    int gi = (int)floorf(gx);
    int gj = (int)floorf(gy);
    float aw = c_aw[bn], ah = c_ah[bn];

    // assigned predicted box at (b, bn, gj, gi)
    int base = ((b * NA + bn) * CH) * PLANE + gj * NW + gi;
    float o0 = out[base];
    float o1 = out[base + PLANE];
    float o2 = out[base + 2 * PLANE];
    float o3 = out[base + 3 * PLANE];
    float px = sigmoidf_fast(o0) + (float)gi;
    float py = sigmoidf_fast(o1) + (float)gj;
    float pw = __expf(o2) * aw;
    float ph = __expf(o3) * ah;

    float gxl = gx - 0.5f * gw, gxr = gx + 0.5f * gw;
    float gyl = gy - 0.5f * gh, gyr = gy + 0.5f * gh;
    float pxl = px - 0.5f * pw, pxr = px + 0.5f * pw;
    float pyl = py - 0.5f * ph, pyr = py + 0.5f * ph;
    float cw = fminf(gxr, pxr) - fmaxf(gxl, pxl);
    float chh = fminf(gyr, pyr) - fmaxf(gyl, pyl);
    float inter = (cw > 0.0f && chh > 0.0f) ? cw * chh : 0.0f;
    float tconf = __fdividef(inter, area_g + pw * ph - inter);

    int cell = (bn * NH + gj) * NW + gi;  // cell id within batch [0, 3920)

    float* r = rec + (b * NT + t) * RECF;
    r[0] = gxl; r[1] = gxr; r[2] = gyl; r[3] = gyr;
    r[4] = area_g;
    r[5] = __int_as_float(cell);
    r[6] = gx - (float)gi;
    r[7] = gy - (float)gj;
    r[8] = __logf(__fdividef(gw, aw));
    r[9] = __logf(__fdividef(gh, ah));
    r[10] = tconf;
    r[11] = cls;
    r[12] = 0.0f; r[13] = 0.0f; r[14] = 0.0f; r[15] = 0.0f;
}

__global__ void loss_kernel(const float* __restrict__ out,
                            const float* __restrict__ rec,
                            float* __restrict__ partials) {
    __shared__ __attribute__((aligned(64))) float sm[NT * RECF];  // 3200 B
    __shared__ float wsum[8];

    int b = blockIdx.y;
    int tid = threadIdx.x;

    // Async-copy this batch's 50x16-float GT table into LDS (CDNA5 async path).
    // 200 lanes x b128 = 3200 bytes.
    if (tid < (NT * RECF) / 4) {
        unsigned long long ga =
            (unsigned long long)(const void*)(rec + b * (NT * RECF) + tid * 4);
        unsigned la = (unsigned)(unsigned long long)(const void*)&sm[tid * 4];
        asm volatile("global_load_async_to_lds_b128 %0, %1, off"
                     :: "v"(la), "v"(ga) : "memory");
    }
    asm volatile("s_wait_asynccnt 0" ::: "memory");
    __syncthreads();

    float acc = 0.0f;
    int cell = blockIdx.x * blockDim.x + tid;
    if (cell < CELLS_PER_B) {
        int a = cell / PLANE;
        int remp = cell - a * PLANE;
        int j = remp / NW;
        int i = remp - j * NW;
        int base = ((b * NA + a) * CH) * PLANE + remp;

        float o0 = out[base];
        float o1 = out[base + PLANE];
        float o2 = out[base + 2 * PLANE];
        float o3 = out[base + 3 * PLANE];
        float o4 = out[base + 4 * PLANE];

        float x = sigmoidf_fast(o0);
        float y = sigmoidf_fast(o1);
        float conf = sigmoidf_fast(o4);
        float pw = __expf(o2) * c_aw[a];
        float ph = __expf(o3) * c_ah[a];
        float px = x + (float)i;
        float py = y + (float)j;
        float pxl = px - 0.5f * pw, pxr = px + 0.5f * pw;
        float pyl = py - 0.5f * ph, pyr = py + 0.5f * ph;
        float ap = pw * ph;

        // max_iou is only compared against SIL_THRESH, so reduce
        // max_t(inter/union) > thr  to  max_t(inter - thr*union) > 0:
        // pure full-rate VALU (fma+max), no per-iteration rcp.
        float over = -1.0f;
        int tm = -1;
#pragma unroll 5
        for (int t = 0; t < NT; ++t) {
            v4f bx = *(const v4f*)&sm[t * RECF];      // xl xr yl yr  (ds_load_b128)
            v2f ae = *(const v2f*)&sm[t * RECF + 4];  // area, cell   (ds_load_b64)
            float cw  = fminf(bx.y, pxr) - fmaxf(bx.x, pxl);
            float chh = fminf(bx.w, pyr) - fmaxf(bx.z, pyl);
            float inter = (cw > 0.0f && chh > 0.0f) ? cw * chh : 0.0f;
            float uni = ap + ae.x - inter;
            over = fmaxf(over, fmaf(-SIL_THRESH, uni, inter));
            if (__float_as_int(ae.y) == cell) tm = t;  // last write wins
        }

        float cm2, tconf;
        if (tm >= 0) {
            const float* r = &sm[tm * RECF];
            // smooth-L1 coord loss (coord_mask == 1 here), x0.5
            float s = 0.0f, d, ad;
            d = x  - r[6]; ad = fabsf(d); s += (ad < 1.0f) ? 0.5f * d * d : ad - 0.5f;
            d = y  - r[7]; ad = fabsf(d); s += (ad < 1.0f) ? 0.5f * d * d : ad - 0.5f;
            d = o2 - r[8]; ad = fabsf(d); s += (ad < 1.0f) ? 0.5f * d * d : ad - 0.5f;
            d = o3 - r[9]; ad = fabsf(d); s += (ad < 1.0f) ? 0.5f * d * d : ad - 0.5f;
            acc += 0.5f * s;
            cm2 = OBJECT_SCALE;
            tconf = r[10];

            // focal class loss, online log-sum-exp over 80 logits
            int tc = (int)r[11];
            const float* cl = out + base + 5 * PLANE;
            float m = -1e30f, ssum = 0.0f, lt = 0.0f;
            for (int c = 0; c < NC; ++c) {
                float v = cl[c * PLANE];
                float mn = fmaxf(m, v);
                ssum = ssum * __expf(m - mn) + __expf(v - mn);
                m = mn;
                if (c == tc) lt = v;
            }
            float logp = lt - (m + __logf(ssum));
            float p = __expf(logp);
            float om = 1.0f - p;
            acc += -(om * om) * logp;
        } else {
            cm2 = (over > 0.0f) ? 0.0f : 1.0f;  // NOOBJECT_SCALE or silenced
            tconf = 0.0f;
        }
        float dc = conf - tconf;
        acc += 0.5f * cm2 * dc * dc;
    }

    // wave32 reduction then cross-wave via LDS (deterministic)
#pragma unroll
    for (int o = 16; o > 0; o >>= 1) acc += __shfl_xor(acc, o, 32);
    int wave = tid >> 5, lane = tid & 31;
    if (lane == 0) wsum[wave] = acc;
    __syncthreads();
    if (tid == 0) {
        float s = 0.0f;
#pragma unroll
        for (int w = 0; w < 8; ++w) s += wsum[w];
        partials[blockIdx.y * gridDim.x + blockIdx.x] = s;
    }
}

__global__ void reduce_kernel(const float* __restrict__ partials,
                              float* __restrict__ outv, int n) {
    __shared__ float smr[256];
    int tid = threadIdx.x;
    float s = 0.0f;
    for (int i = tid; i < n; i += 256) s += partials[i];
    smr[tid] = s;
    __syncthreads();
    for (int o = 128; o > 0; o >>= 1) {
        if (tid < o) smr[tid] += smr[tid + o];
        __syncthreads();
    }
    if (tid == 0) outv[0] = smr[0];
}

extern "C" void kernel_launch(void* const* d_in, const int* in_sizes, int n_in,
                              void* d_out, int out_size, void* d_ws, size_t ws_size,
                              hipStream_t stream) {
    const float* output = (const float*)d_in[0];
    const float* target = (const float*)d_in[1];

    float* rec = (float*)d_ws;                       // NB*NT*16 floats = 409600 B
    float* partials = rec + (size_t)NB * NT * RECF;  // 16*NB floats

    build_gt_kernel<<<NB, 64, 0, stream>>>(output, target, rec);

    dim3 grid((CELLS_PER_B + 255) / 256, NB);
    loss_kernel<<<grid, 256, 0, stream>>>(output, rec, partials);

    int nblocks = ((CELLS_PER_B + 255) / 256) * NB;
    reduce_kernel<<<1, 256, 0, stream>>>(partials, (float*)d_out, nblocks);
}